// AttentionLayer_532575944777
// MI455X (gfx1250) — compile-verified
//
#include <hip/hip_runtime.h>

// ---------------------------------------------------------------------------
// Relation-aware self-attention (Shaw et al.) for MI455X / gfx1250.
// B=16, S=1024, DIN=D=64. All math in exact fp32 via V_WMMA_F32_16X16X4_F32.
//
// Pipeline:
//   k1: q,k,v = x @ {Wq,Wk,Wv}                       (WMMA, trivial cost)
//   k2: fused scores + exp + unnormalized z          (WMMA, streams alphas once)
//       grid = (64 i-tiles, 4 j-chunks), 16 waves/WG.
//       wave w = "batch w" for q.kT and P@v GEMMs (M = 16 query rows)
//       wave w = "query i0+w" for q.alphaK and P@alphaV GEMMs (M = 16 batches)
//       16x16 score/prob tiles exchanged through padded LDS.
//   k3: z = (sum of chunk partials) / rowsum; emit final rowsums
//   k4: attn /= rowsum (in place, float4)
// ---------------------------------------------------------------------------

typedef __attribute__((ext_vector_type(2))) float v2f;
typedef __attribute__((ext_vector_type(8))) float v8f;

#define S_LEN   1024
#define DMODEL  64
#define NBATCH  16
#define JCHUNK  256
#define NCHUNK  4
#define INV_SQRT_D 0.125f

__device__ __forceinline__ v8f wmma_f32_16x16x4(v2f a, v2f b, v8f c) {
  // D(16x16,f32) = A(16x4,f32) * B(4x16,f32) + C
  return __builtin_amdgcn_wmma_f32_16x16x4_f32(
      /*neg_a=*/false, a, /*neg_b=*/false, b,
      /*c_mod=*/(short)0, c, /*reuse_a=*/false, /*reuse_b=*/false);
}

__device__ __forceinline__ v8f v8f_zero() {
  v8f r;
#pragma unroll
  for (int i = 0; i < 8; ++i) r[i] = 0.0f;
  return r;
}

// ---------------------------------------------------------------------------
// Kernel 1: q/k/v projections. One wave per 16x16 output tile.
// grid = (B*S/16, 3 matrices), block = 128 (4 waves = 4 n-tiles of D=64).
// ---------------------------------------------------------------------------
__global__ __launch_bounds__(128) void qkv_kernel(
    const float* __restrict__ x, const float* __restrict__ Wq,
    const float* __restrict__ Wk, const float* __restrict__ Wv,
    float* __restrict__ q, float* __restrict__ k, float* __restrict__ v)
{
  const int rowtile = blockIdx.x;            // 0..1023  (flat b*S+s rows /16)
  const int mat     = blockIdx.y;            // 0..2
  const int wave    = threadIdx.x >> 5;      // n-tile 0..3
  const int lane    = threadIdx.x & 31;
  const int m       = lane & 15;
  const int h       = lane >> 4;

  const float* W   = (mat == 0) ? Wq : (mat == 1) ? Wk : Wv;
  float*       out = (mat == 0) ? q  : (mat == 1) ? k  : v;

  const int r0 = rowtile * 16;
  const int n0 = wave * 16;

  v8f acc = v8f_zero();
#pragma unroll
  for (int kk = 0; kk < 16; ++kk) {
    const int dd = 4 * kk + 2 * h;
    v2f a = *(const v2f*)(x + (r0 + m) * DMODEL + dd);      // A[m, dd..dd+1]
    v2f b;
    b.x = W[(dd)     * DMODEL + n0 + m];                    // B[dd,   n]
    b.y = W[(dd + 1) * DMODEL + n0 + m];                    // B[dd+1, n]
    acc = wmma_f32_16x16x4(a, b, acc);
  }
#pragma unroll
  for (int g = 0; g < 8; ++g)
    out[(r0 + g + 8 * h) * DMODEL + n0 + m] = acc[g];
}

// ---------------------------------------------------------------------------
// Kernel 2: fused scores + exp + unnormalized z, one j-chunk per WG.
// ---------------------------------------------------------------------------
__global__ __launch_bounds__(512) void attn_core_kernel(
    const float* __restrict__ q, const float* __restrict__ k,
    const float* __restrict__ v, const float* __restrict__ aK,
    const float* __restrict__ aV, float* __restrict__ attn,
    float* __restrict__ zpart, float* __restrict__ sumspart)
{
  const int itile = blockIdx.x;              // 0..63
  const int chunk = blockIdx.y;              // 0..3
  const int i0    = itile * 16;
  const int jbase = chunk * JCHUNK;
  const int w     = threadIdx.x >> 5;        // wave id 0..15
  const int lane  = threadIdx.x & 31;
  const int m     = lane & 15;
  const int h     = lane >> 4;
  const int iG    = i0 + w;                  // this wave's query row (alpha GEMMs)

  // Padded LDS: strides chosen for conflict-free strided reads.
  __shared__ float pL [16][17][18];          // score -> prob tiles  p[b][i][j]
  __shared__ float s2L[16][17][18];          // alpha-score tiles    s2[i][b][j]
  __shared__ float exL[16][16][17];          // z combine exchange (per d-tile)

  // Resident A-fragments (K = 64 -> 16 frags of 16x4).
  v2f QbA[16];   // A[m=i_local, d]  for batch b = w      (q.kT, P@v GEMMs)
  v2f QiA[16];   // A[m=batch,   d]  for query i = iG     (alpha GEMMs)
#pragma unroll
  for (int kk = 0; kk < 16; ++kk) {
    const int dd = 4 * kk + 2 * h;
    QbA[kk] = *(const v2f*)(q + (w * S_LEN + i0 + m) * DMODEL + dd);
    QiA[kk] = *(const v2f*)(q + (m * S_LEN + iG)     * DMODEL + dd);
  }

  v8f z1[4], z2[4];                          // 4 d-tiles of 16 cols each
#pragma unroll
  for (int t = 0; t < 4; ++t) { z1[t] = v8f_zero(); z2[t] = v8f_zero(); }
  float rowSum = 0.0f;

  for (int jt = 0; jt < JCHUNK / 16; ++jt) {
    const int j0 = jbase + jt * 16;

    // ---- scores: s1[b=w][i][j] = q_b . k_b^T ; s2[i=iG][b][j] = q_i . aK_i^T
    v8f s1 = v8f_zero();
    v8f s2 = v8f_zero();
#pragma unroll
    for (int kk = 0; kk < 16; ++kk) {
      const int dd = 4 * kk + 2 * h;
      v2f bk = *(const v2f*)(k + (w * S_LEN + j0 + m) * DMODEL + dd);
      s1 = wmma_f32_16x16x4(QbA[kk], bk, s1);
      v2f ba = __builtin_nontemporal_load(
          (const v2f*)(aK + ((size_t)iG * S_LEN + j0 + m) * DMODEL + dd));
      s2 = wmma_f32_16x16x4(QiA[kk], ba, s2);
    }
#pragma unroll
    for (int g = 0; g < 8; ++g) {
      pL [w][g + 8 * h][m] = s1[g];
      s2L[w][g + 8 * h][m] = s2[g];
    }
    __syncthreads();

    // ---- exp: wave w = batch; lane handles row i=m, cols j = 8h..8h+7
    float pv[8];
    float ps = 0.0f;
#pragma unroll
    for (int jj = 0; jj < 8; ++jj) {
      const int j = 8 * h + jj;
      const float sc = (pL[w][m][j] + s2L[m][w][j]) * INV_SQRT_D;
      const float e  = __expf(sc);           // |sc| small: safe w/o max-sub
      pv[jj] = e;
      ps += e;
    }
#pragma unroll
    for (int jj = 0; jj < 8; ++jj) pL[w][m][8 * h + jj] = pv[jj];
    rowSum += ps + __shfl_xor(ps, 16, 32);   // pair lanes (m, m+16) share a row

    // unnormalized probs out (normalized later by kernel 4)
    float* arow = attn + ((size_t)(w * S_LEN + i0 + m) * S_LEN + j0 + 8 * h);
    ((float4*)arow)[0] = make_float4(pv[0], pv[1], pv[2], pv[3]);
    ((float4*)arow)[1] = make_float4(pv[4], pv[5], pv[6], pv[7]);
    __syncthreads();

    // ---- z accumulation (K = 16 j's -> 4 k-steps)
    v2f pA[4], pE[4];
#pragma unroll
    for (int kk = 0; kk < 4; ++kk) {
      const int jj0 = 4 * kk + 2 * h;
      pA[kk].x = pL[w][m][jj0];  pA[kk].y = pL[w][m][jj0 + 1];  // A[m=i, k=j]
      pE[kk].x = pL[m][w][jj0];  pE[kk].y = pL[m][w][jj0 + 1];  // A[m=b, k=j]
    }
#pragma unroll
    for (int dt = 0; dt < 4; ++dt) {
#pragma unroll
      for (int kk = 0; kk < 4; ++kk) {
        const int jrow = j0 + 4 * kk + 2 * h;
        v2f bv, bav;
        bv.x  = v[(w * S_LEN + jrow)     * DMODEL + dt * 16 + m];
        bv.y  = v[(w * S_LEN + jrow + 1) * DMODEL + dt * 16 + m];
        z1[dt] = wmma_f32_16x16x4(pA[kk], bv, z1[dt]);
        bav.x = __builtin_nontemporal_load(
            aV + ((size_t)iG * S_LEN + jrow)     * DMODEL + dt * 16 + m);
        bav.y = __builtin_nontemporal_load(
            aV + ((size_t)iG * S_LEN + jrow + 1) * DMODEL + dt * 16 + m);
        z2[dt] = wmma_f32_16x16x4(pE[kk], bav, z2[dt]);
      }
    }
    __syncthreads();
  }

  // ---- combine z1 (wave=batch slice) + z2 (wave=query slice) per d-tile
  for (int dt = 0; dt < 4; ++dt) {
#pragma unroll
    for (int g = 0; g < 8; ++g)
      exL[w][g + 8 * h][m] = z1[dt][g];            // [b][i][d]
    __syncthreads();
#pragma unroll
    for (int g = 0; g < 8; ++g)
      exL[g + 8 * h][w][m] += z2[dt][g];           // add [b][i=w][d]
    __syncthreads();
    for (int off = threadIdx.x; off < 16 * 16 * 16; off += 512) {
      const int b  = off >> 8;
      const int ii = (off >> 4) & 15;
      const int d  = off & 15;
      zpart[(((size_t)(chunk * 16 + b)) * S_LEN + i0 + ii) * DMODEL
            + dt * 16 + d] = exL[b][ii][d];
    }
    __syncthreads();
  }

  if (lane < 16)
    sumspart[(chunk * 16 + w) * S_LEN + i0 + lane] = rowSum;
}

// ---------------------------------------------------------------------------
// Kernel 3: z = sum(chunk partials)/rowsum; emit final rowsums.
// ---------------------------------------------------------------------------
__global__ __launch_bounds__(256) void z_finalize_kernel(
    const float* __restrict__ zpart, const float* __restrict__ sumspart,
    float* __restrict__ z_out, float* __restrict__ sumsF)
{
  const int idx = blockIdx.x * 256 + threadIdx.x;   // < 16*1024*64
  const int b = idx >> 16;
  const int s = (idx >> 6) & (S_LEN - 1);
  const int d = idx & 63;
  float sum = 0.0f, zv = 0.0f;
#pragma unroll
  for (int c = 0; c < NCHUNK; ++c) {
    sum += sumspart[(c * 16 + b) * S_LEN + s];
    zv  += zpart[(((size_t)(c * 16 + b)) * S_LEN + s) * DMODEL + d];
  }
  z_out[idx] = zv / sum;
  if (d == 0) sumsF[b * S_LEN + s] = sum;
}

// ---------------------------------------------------------------------------
// Kernel 4: attn /= rowsum, in place, float4.
// ---------------------------------------------------------------------------
__global__ __launch_bounds__(256) void attn_norm_kernel(
    float* __restrict__ attn, const float* __restrict__ sumsF)
{
  const int t   = blockIdx.x * 256 + threadIdx.x;   // < 16M/4
  const int row = t >> 8;                           // b*S + i (256 float4/row)
  const float inv = 1.0f / sumsF[row];
  float4* a4 = (float4*)attn;
  float4 p = a4[t];
  p.x *= inv; p.y *= inv; p.z *= inv; p.w *= inv;
  a4[t] = p;
}

// ---------------------------------------------------------------------------
extern "C" void kernel_launch(void* const* d_in, const int* in_sizes, int n_in,
                              void* d_out, int out_size, void* d_ws, size_t ws_size,
                              hipStream_t stream)
{
  const float* x  = (const float*)d_in[0];
  const float* Wq = (const float*)d_in[1];
  const float* Wk = (const float*)d_in[2];
  const float* Wv = (const float*)d_in[3];
  const float* aK = (const float*)d_in[4];
  const float* aV = (const float*)d_in[5];

  float* ws       = (float*)d_ws;
  float* q        = ws;                 // 1,048,576 floats
  float* k        = ws + 1048576;       // 1,048,576
  float* v        = ws + 2097152;       // 1,048,576
  float* zpart    = ws + 3145728;       // 4,194,304 (4 chunks x B x S x D)
  float* sumspart = ws + 7340032;       //    65,536 (4 chunks x B x S)
  float* sumsF    = ws + 7405568;       //    16,384 (B x S)
  // total ws use: ~28.3 MB

  float* z_out = (float*)d_out;         // z: 16*1024*64
  float* attn  = z_out + 1048576;       // attn: 16*1024*1024

  qkv_kernel<<<dim3(1024, 3), 128, 0, stream>>>(x, Wq, Wk, Wv, q, k, v);
  attn_core_kernel<<<dim3(64, NCHUNK), 512, 0, stream>>>(
      q, k, v, aK, aV, attn, zpart, sumspart);
  z_finalize_kernel<<<4096, 256, 0, stream>>>(zpart, sumspart, z_out, sumsF);
  attn_norm_kernel<<<16384, 256, 0, stream>>>(attn, sumsF);
}